// quantizer_49314814492727
// MI455X (gfx1250) — compile-verified
//
#include <hip/hip_runtime.h>

typedef __attribute__((ext_vector_type(2))) float v2f;
typedef __attribute__((ext_vector_type(8))) float v8f;

#define K_CODES 512
#define DIMS 64
#define TILES_PER_WAVE 2
#define WAVES_PER_BLOCK 8
#define BLOCK_THREADS 256
#define NUM_ROWS (16 * 4096)
#define NUM_TILES (NUM_ROWS / 16)

__global__ __launch_bounds__(BLOCK_THREADS)
void vq_argmin_wmma_kernel(const float* __restrict__ encoded,
                           const float* __restrict__ embedding,
                           float* __restrict__ out)
{
    extern __shared__ float smem[];
    float* lds_emb  = smem;                    // 512 x 64 floats = 128 KB
    float* lds_bias = smem + K_CODES * DIMS;   // 512 floats

    const int t = threadIdx.x;

    // ---- Stage codebook into LDS (coalesced float4) ----
    {
        const float4* src = (const float4*)embedding;
        float4*       dst = (float4*)lds_emb;
        #pragma unroll
        for (int i = 0; i < (K_CODES * DIMS / 4) / BLOCK_THREADS; ++i)
            dst[t + i * BLOCK_THREADS] = src[t + i * BLOCK_THREADS];
    }
    __syncthreads();

    // ---- bias[c] = -0.5 * ||c||^2 (folded into WMMA accumulator init) ----
    for (int c = t; c < K_CODES; c += BLOCK_THREADS) {
        float s = 0.f;
        #pragma unroll
        for (int d = 0; d < DIMS; ++d) { float v = lds_emb[c * DIMS + d]; s += v * v; }
        lds_bias[c] = -0.5f * s;
    }
    __syncthreads();

    const int lane = t & 31;
    const int wave = t >> 5;
    const int r    = lane & 15;   // A-row / B-code / C-column within 16x16 tile
    const int kh   = lane >> 4;   // K-half selector for f32 16x16x4 fragments

    const int tile0 = (blockIdx.x * WAVES_PER_BLOCK + wave) * TILES_PER_WAVE;

    // ---- Load resident A fragments: A reg pair = x[row][4*kk + 2*kh .. +1] ----
    v2f A[TILES_PER_WAVE][16];
    #pragma unroll
    for (int rt = 0; rt < TILES_PER_WAVE; ++rt) {
        const float* xrow = encoded + ((size_t)(tile0 + rt) * 16 + r) * DIMS + 2 * kh;
        #pragma unroll
        for (int kk = 0; kk < 16; ++kk)
            A[rt][kk] = *(const v2f*)(xrow + 4 * kk);
    }

    float bestv[TILES_PER_WAVE][8];
    int   besti[TILES_PER_WAVE][8];
    #pragma unroll
    for (int rt = 0; rt < TILES_PER_WAVE; ++rt)
        #pragma unroll
        for (int j = 0; j < 8; ++j) { bestv[rt][j] = -3.402823466e38f; besti[rt][j] = 0; }

    // ---- 32 code tiles: 16 chained f32 WMMAs each, acc seeded with bias ----
    for (int ct = 0; ct < 32; ++ct) {
        const int   cb   = ct * 16;
        const float bias = lds_bias[cb + r];
        v8f acc[TILES_PER_WAVE];
        #pragma unroll
        for (int rt = 0; rt < TILES_PER_WAVE; ++rt) {
            v8f a = {bias, bias, bias, bias, bias, bias, bias, bias};
            acc[rt] = a;
        }
        const float* bp = lds_emb + (size_t)(cb + r) * DIMS + 2 * kh;
        #pragma unroll
        for (int kk = 0; kk < 16; ++kk) {
            v2f b = *(const v2f*)(bp + 4 * kk);   // ds_load_b64, shared by both row tiles
            #pragma unroll
            for (int rt = 0; rt < TILES_PER_WAVE; ++rt)
                acc[rt] = __builtin_amdgcn_wmma_f32_16x16x4_f32(
                    false, A[rt][kk], false, b, (short)0, acc[rt], false, false);
        }
        const int myidx = cb + r;
        #pragma unroll
        for (int rt = 0; rt < TILES_PER_WAVE; ++rt)
            #pragma unroll
            for (int j = 0; j < 8; ++j)
                if (acc[rt][j] > bestv[rt][j]) { bestv[rt][j] = acc[rt][j]; besti[rt][j] = myidx; }
    }

    // ---- Cross-lane argmax over the 16 columns of each half-wave ----
    #pragma unroll
    for (int rt = 0; rt < TILES_PER_WAVE; ++rt)
        #pragma unroll
        for (int j = 0; j < 8; ++j) {
            float v  = bestv[rt][j];
            int   id = besti[rt][j];
            #pragma unroll
            for (int s = 1; s < 16; s <<= 1) {
                float ov = __shfl_xor(v, s, 16);
                int   oi = __shfl_xor(id, s, 16);
                // max score; lower index wins ties (matches argmin-first semantics)
                if (ov > v || (ov == v && oi < id)) { v = ov; id = oi; }
            }
            besti[rt][j] = id;
        }

    // ---- Gather winning code rows from LDS, coalesced float4 stores ----
    #pragma unroll
    for (int rt = 0; rt < TILES_PER_WAVE; ++rt)
        #pragma unroll
        for (int j = 0; j < 8; ++j) {
            const int m   = kh * 8 + j;              // row covered by this lane half
            const int idx = besti[rt][j];
            const float4 q = *(const float4*)(lds_emb + (size_t)idx * DIMS + r * 4);
            *(float4*)(out + ((size_t)(tile0 + rt) * 16 + m) * DIMS + r * 4) = q;
        }
}

extern "C" void kernel_launch(void* const* d_in, const int* in_sizes, int n_in,
                              void* d_out, int out_size, void* d_ws, size_t ws_size,
                              hipStream_t stream) {
    const float* encoded   = (const float*)d_in[0];
    const float* embedding = (const float*)d_in[1];
    float*       out       = (float*)d_out;

    const size_t shmem = (size_t)(K_CODES * DIMS + K_CODES) * sizeof(float); // ~130 KB (LDS is 320 KB/WGP)
    (void)hipFuncSetAttribute((const void*)vq_argmin_wmma_kernel,
                              hipFuncAttributeMaxDynamicSharedMemorySize, (int)shmem);

    const int grid = NUM_TILES / (WAVES_PER_BLOCK * TILES_PER_WAVE); // 4096/16 = 256 blocks
    vq_argmin_wmma_kernel<<<dim3(grid), dim3(BLOCK_THREADS), shmem, stream>>>(encoded, embedding, out);
}